// GAT_Encoder_38053410242786
// MI455X (gfx1250) — compile-verified
//
#include <hip/hip_runtime.h>
#include <hip/hip_bf16.h>

typedef __attribute__((ext_vector_type(2))) float v2f;
typedef __attribute__((ext_vector_type(8))) float v8f;

#define N_NODES   150000
#define USER_CNT  100000
#define LATENT    128
#define N_EDGES_C 2000000
#define BATCH_C   4096
#define NEG_SLOPE 0.2f

// ---------------------------------------------------------------------------
// GEMM: H[N x 128] = X[N x 128] @ W[128 x 128] using fp32 WMMA 16x16x4.
// One wave per 16-row stripe; covers all 8 column tiles (reuses A in regs).
// ---------------------------------------------------------------------------
__global__ __launch_bounds__(256) void gemm_h_kernel(
    const float* __restrict__ X, const float* __restrict__ W,
    float* __restrict__ H, int nrows) {
  const int wave = (int)((blockIdx.x * blockDim.x + threadIdx.x) >> 5);
  const int lane = (int)(threadIdx.x & 31);
  const int row0 = wave * 16;
  if (row0 >= nrows) return;                 // wave-uniform: EXEC stays all-1s
  const int half = lane >> 4;                // 0: lanes 0-15, 1: lanes 16-31
  const int l16  = lane & 15;
  const int m    = row0 + l16;               // A row owned by this lane

  v8f acc[8] = {};                           // 8 column tiles of C/D
#pragma unroll 4
  for (int k0 = 0; k0 < 128; k0 += 4) {
    const int ka = k0 + half * 2;            // A 16x4 layout: VGPR0={K0,K2}, VGPR1={K1,K3}
    v2f a;
    a.x = X[(size_t)m * 128 + ka];
    a.y = X[(size_t)m * 128 + ka + 1];
#pragma unroll
    for (int t = 0; t < 8; ++t) {
      const int n = t * 16 + l16;            // B 4x16 layout mirrors A (row across lanes)
      v2f b;
      b.x = W[ka * 128 + n];
      b.y = W[(ka + 1) * 128 + n];
      acc[t] = __builtin_amdgcn_wmma_f32_16x16x4_f32(
          false, a, false, b, (short)0, acc[t], false, false);
    }
  }
  // C/D 16x16 layout: acc[t][r] -> row (row0 + r + 8*half), col (16t + l16)
#pragma unroll
  for (int t = 0; t < 8; ++t) {
#pragma unroll
    for (int r = 0; r < 8; ++r) {
      H[(size_t)(row0 + r + 8 * half) * 128 + t * 16 + l16] = acc[t][r];
    }
  }
}

// ---------------------------------------------------------------------------
// logits: ls[i] = h[i]·a_s, ld[i] = h[i]·a_d  (one wave per node)
// ---------------------------------------------------------------------------
__global__ __launch_bounds__(256) void logits_kernel(
    const float* __restrict__ H, const float* __restrict__ a_s,
    const float* __restrict__ a_d, float* __restrict__ ls,
    float* __restrict__ ld, int n) {
  const int wave = (int)((blockIdx.x * blockDim.x + threadIdx.x) >> 5);
  const int lane = (int)(threadIdx.x & 31);
  if (wave >= n) return;
  const float* h = H + (size_t)wave * 128;
  float s = 0.f, d = 0.f;
#pragma unroll
  for (int i = 0; i < 4; ++i) {
    const float hv = h[lane + 32 * i];
    s += hv * a_s[lane + 32 * i];
    d += hv * a_d[lane + 32 * i];
  }
#pragma unroll
  for (int off = 16; off > 0; off >>= 1) {
    s += __shfl_down(s, off, 32);
    d += __shfl_down(d, off, 32);
  }
  if (lane == 0) { ls[wave] = s; ld[wave] = d; }
}

// ---------------------------------------------------------------------------
// init: out = bias (broadcast), seg_max = -inf, denom = 0
// ---------------------------------------------------------------------------
__global__ void init_kernel(float* __restrict__ out,
                            const float* __restrict__ bias,
                            float* __restrict__ smax,
                            float* __restrict__ den, int n) {
  const int i = (int)(blockIdx.x * blockDim.x + threadIdx.x);
  if (i >= n * 128) return;
  out[i] = bias[i & 127];
  if ((i & 127) == 0) {
    const int node = i >> 7;
    smax[node] = -__builtin_inff();
    den[node]  = 0.f;
  }
}

__device__ __forceinline__ float leaky(float x) {
  return x > 0.f ? x : NEG_SLOPE * x;
}

// float atomic max via monotonic signed/unsigned int mapping
__device__ __forceinline__ void atomicMaxF32(float* addr, float v) {
  if (v >= 0.f) atomicMax((int*)addr, __float_as_int(v));
  else          atomicMin((unsigned int*)addr, __float_as_uint(v));
}

__device__ __forceinline__ void edge_endpoints(const int* __restrict__ ei,
                                               int t, int E, int* s, int* d) {
  if (t < E) { *s = ei[t]; *d = ei[E + t]; }       // edge_index rows 0/1
  else       { *s = *d = t - E; }                  // appended self loop
}

// ---------------------------------------------------------------------------
// pass 1: seg_max over dst
// ---------------------------------------------------------------------------
__global__ void edge_max_kernel(const int* __restrict__ ei,
                                const float* __restrict__ ls,
                                const float* __restrict__ ld,
                                float* __restrict__ smax, int E, int n) {
  const int t = (int)(blockIdx.x * blockDim.x + threadIdx.x);
  if (t >= E + n) return;
  int s, d; edge_endpoints(ei, t, E, &s, &d);
  atomicMaxF32(&smax[d], leaky(ls[s] + ld[d]));
}

// ---------------------------------------------------------------------------
// pass 2: w = exp(e - max[dst]); denom[dst] += w
// ---------------------------------------------------------------------------
__global__ void edge_sum_kernel(const int* __restrict__ ei,
                                const float* __restrict__ ls,
                                const float* __restrict__ ld,
                                const float* __restrict__ smax,
                                float* __restrict__ den,
                                float* __restrict__ wtmp, int E, int n) {
  const int t = (int)(blockIdx.x * blockDim.x + threadIdx.x);
  if (t >= E + n) return;
  int s, d; edge_endpoints(ei, t, E, &s, &d);
  const float w = expf(leaky(ls[s] + ld[d]) - smax[d]);
  wtmp[t] = w;
  atomicAdd(&den[d], w);
}

// ---------------------------------------------------------------------------
// pass 3: out[dst] += alpha * h[src]   (one wave per edge, float4 per lane)
// ---------------------------------------------------------------------------
__global__ __launch_bounds__(256) void scatter_kernel(
    const int* __restrict__ ei, const float* __restrict__ H,
    const float* __restrict__ den, const float* __restrict__ wtmp,
    float* __restrict__ out, int E, int n) {
  const int gw   = (int)((blockIdx.x * blockDim.x + threadIdx.x) >> 5);
  const int lane = (int)(threadIdx.x & 31);
  if (gw >= E + n) return;
  int s, d; edge_endpoints(ei, gw, E, &s, &d);
  const float alpha = wtmp[gw] / den[d];
  const float4 hv = ((const float4*)(H + (size_t)s * 128))[lane];
  float* od = out + (size_t)d * 128 + lane * 4;
  atomicAdd(od + 0, hv.x * alpha);
  atomicAdd(od + 1, hv.y * alpha);
  atomicAdd(od + 2, hv.z * alpha);
  atomicAdd(od + 3, hv.w * alpha);
}

// ---------------------------------------------------------------------------
// final gather: out = concat(x[user], x[USER_CNT + item]) flat
// ---------------------------------------------------------------------------
__global__ void gather_kernel(const float* __restrict__ X,
                              const int* __restrict__ user,
                              const int* __restrict__ item,
                              float* __restrict__ out, int B) {
  const int t = (int)(blockIdx.x * blockDim.x + threadIdx.x);
  const int tot = 2 * B * 128;
  if (t >= tot) return;
  const int half = t / (B * 128);
  const int r = (t - half * B * 128) >> 7;
  const int c = t & 127;
  const int node = (half == 0) ? user[r] : (USER_CNT + item[r]);
  out[t] = X[(size_t)node * 128 + c];
}

extern "C" void kernel_launch(void* const* d_in, const int* in_sizes, int n_in,
                              void* d_out, int out_size, void* d_ws, size_t ws_size,
                              hipStream_t stream) {
  (void)in_sizes; (void)n_in; (void)out_size; (void)ws_size;
  const int*   ei   = (const int*)d_in[0];     // [2, E]
  const int*   user = (const int*)d_in[1];     // [B]
  const int*   item = (const int*)d_in[2];     // [B]
  const float* uemb = (const float*)d_in[3];   // [100000,128]
  const float* iemb = (const float*)d_in[4];   // [50000,128]
  const float* W    = (const float*)d_in[5];   // [3,128,128]
  const float* a_s  = (const float*)d_in[6];   // [3,128]
  const float* a_d  = (const float*)d_in[7];   // [3,128]
  const float* bias = (const float*)d_in[8];   // [3,128]

  const int N = N_NODES, E = N_EDGES_C, B = BATCH_C;
  const int Etot = E + N;
  const size_t ND = (size_t)N * 128;

  float* ws   = (float*)d_ws;
  float* xbuf = ws;                 // [N,128]
  float* hbuf = xbuf + ND;          // [N,128]
  float* obuf = hbuf + ND;          // [N,128]
  float* ls   = obuf + ND;          // [N]
  float* ldv  = ls + N;             // [N]
  float* smax = ldv + N;            // [N]
  float* den  = smax + N;           // [N]
  float* wtmp = den + N;            // [Etot]

  // x0 = concat(user_emb, item_emb)
  hipMemcpyAsync(xbuf, uemb, (size_t)USER_CNT * 128 * sizeof(float),
                 hipMemcpyDeviceToDevice, stream);
  hipMemcpyAsync(xbuf + (size_t)USER_CNT * 128, iemb,
                 (size_t)(N - USER_CNT) * 128 * sizeof(float),
                 hipMemcpyDeviceToDevice, stream);

  float* xin = xbuf;
  float* xout = obuf;
  for (int l = 0; l < 3; ++l) {
    gemm_h_kernel<<<(N / 16 + 7) / 8, 256, 0, stream>>>(xin, W + l * 128 * 128, hbuf, N);
    logits_kernel<<<(N + 7) / 8, 256, 0, stream>>>(hbuf, a_s + l * 128, a_d + l * 128,
                                                   ls, ldv, N);
    init_kernel<<<((size_t)N * 128 + 255) / 256, 256, 0, stream>>>(
        xout, bias + l * 128, smax, den, N);
    edge_max_kernel<<<(Etot + 255) / 256, 256, 0, stream>>>(ei, ls, ldv, smax, E, N);
    edge_sum_kernel<<<(Etot + 255) / 256, 256, 0, stream>>>(ei, ls, ldv, smax, den,
                                                            wtmp, E, N);
    scatter_kernel<<<(Etot + 7) / 8, 256, 0, stream>>>(ei, hbuf, den, wtmp, xout, E, N);
    float* tmp = xin; xin = xout; xout = tmp;
  }
  gather_kernel<<<(2 * B * 128 + 255) / 256, 256, 0, stream>>>(xin, user, item,
                                                               (float*)d_out, B);
}